// InterAgg_27642409517102
// MI455X (gfx1250) — compile-verified
//
#include <hip/hip_runtime.h>

#define B_NODES 8192
#define FEAT 128
#define EMBED 128
#define DEG 32
#define KSEL 16

typedef float v2f __attribute__((ext_vector_type(2)));
typedef float v8f __attribute__((ext_vector_type(8)));

// ---------------------------------------------------------------------------
// Kernel 0: gather self features, compute center scores [B,2].
// grid = B, block = 128 (one thread per feature column).
// ---------------------------------------------------------------------------
__global__ void self_gather_scores(const float* __restrict__ feat,
                                   const int* __restrict__ nodes,
                                   const float* __restrict__ Wclf,
                                   const float* __restrict__ bclf,
                                   float* __restrict__ selfF,
                                   float* __restrict__ cs0,
                                   float* __restrict__ scores_out) {
  __shared__ float r0[128];
  __shared__ float r1[128];
  const int b = blockIdx.x;
  const int c = threadIdx.x;                 // 0..127
  const size_t row = (size_t)nodes[b] * FEAT;
  const float v = feat[row + c];
  selfF[(size_t)b * FEAT + c] = v;
  r0[c] = v * Wclf[2 * c];                   // W_clf[c][0]
  r1[c] = v * Wclf[2 * c + 1];               // W_clf[c][1]
  __syncthreads();
  for (int s = 64; s > 0; s >>= 1) {
    if (c < s) { r0[c] += r0[c + s]; r1[c] += r1[c + s]; }
    __syncthreads();
  }
  if (c == 0) {
    const float s0 = r0[0] + bclf[0];
    const float s1 = r1[0] + bclf[1];
    cs0[b] = s0;
    scores_out[2 * b]     = s0;
    scores_out[2 * b + 1] = s1;
  }
}

// ---------------------------------------------------------------------------
// Kernel 1: per (center, relation): gather 32 neighbor rows (coalesced float4),
// fused class-0 score, rank-based top-K=16 selection (stable ties), mean agg.
// grid = (B, 3), block = 256.
// ---------------------------------------------------------------------------
__global__ void relation_agg(const float* __restrict__ feat,
                             const int* __restrict__ n1,
                             const int* __restrict__ n2,
                             const int* __restrict__ n3,
                             const float* __restrict__ Wclf,
                             const float* __restrict__ bclf,
                             const float* __restrict__ cs0,
                             float* __restrict__ aggOut) {
  __shared__ float F[DEG * FEAT];  // 16 KB neighbor tile
  __shared__ float sc[DEG];
  __shared__ float dif[DEG];
  __shared__ int   keep[DEG];

  const int b = blockIdx.x;
  const int r = blockIdx.y;
  const int* __restrict__ neigh = (r == 0) ? n1 : (r == 1) ? n2 : n3;
  const int t = threadIdx.x;                 // 0..255

  if (t < DEG) sc[t] = 0.0f;
  __syncthreads();

  const int j  = t >> 3;                     // neighbor 0..31
  const int l8 = t & 7;                      // 8 threads per row
  const size_t row = (size_t)neigh[b * DEG + j] * FEAT;
  float part = 0.0f;
#pragma unroll
  for (int q = 0; q < 4; ++q) {
    const int col = (q * 8 + l8) * 4;        // consecutive threads -> consecutive float4
    const float4 v = *(const float4*)(feat + row + col);
    *(float4*)(F + j * FEAT + col) = v;
    part += v.x * Wclf[2 * col]
          + v.y * Wclf[2 * (col + 1)]
          + v.z * Wclf[2 * (col + 2)]
          + v.w * Wclf[2 * (col + 3)];
  }
  atomicAdd(&sc[j], part);                   // ds_add_f32 reduce per neighbor
  __syncthreads();

  const float c0 = cs0[b];
  if (t < DEG) dif[t] = fabsf(sc[t] + bclf[0] - c0);
  __syncthreads();

  if (t < DEG) {                             // rank = #(strictly smaller) + stable tie-break
    const float d = dif[t];
    int rank = 0;
    for (int j2 = 0; j2 < DEG; ++j2) {
      const float d2 = dif[j2];
      rank += (d2 < d) || (d2 == d && j2 < t);
    }
    keep[t] = (rank < KSEL) ? 1 : 0;
  }
  __syncthreads();

  if (t < FEAT) {                            // mean over the 16 kept rows
    float s = 0.0f;
#pragma unroll
    for (int j2 = 0; j2 < DEG; ++j2)
      if (keep[j2]) s += F[j2 * FEAT + t];
    aggOut[((size_t)r * B_NODES + b) * FEAT + t] = s * (1.0f / KSEL);
  }
}

// ---------------------------------------------------------------------------
// Kernel 2: E_r = relu([self || agg_r] @ W_intra_r), K=256, fp32 WMMA.
// grid = (B/16, 3), block = 256 = 8 waves; wave w owns N-tile w (16 cols).
// ---------------------------------------------------------------------------
__global__ void intra_gemm(const float* __restrict__ selfF,
                           const float* __restrict__ aggBase,
                           const float* __restrict__ Wi1,
                           const float* __restrict__ Wi2,
                           const float* __restrict__ Wi3,
                           float* __restrict__ Ebase) {
  constexpr int K = 2 * FEAT;                // 256
  constexpr int LDA = K + 4;                 // pad -> conflict-free f32 WMMA A reads
  __shared__ float As[16 * LDA];

  const int mtile = blockIdx.x;
  const int r = blockIdx.y;
  const float* __restrict__ W = (r == 0) ? Wi1 : (r == 1) ? Wi2 : Wi3;
  const float* __restrict__ agg = aggBase + (size_t)r * B_NODES * FEAT;
  float* __restrict__ E = Ebase + (size_t)r * B_NODES * EMBED;

  const int t = threadIdx.x;
  for (int i = t; i < 16 * K; i += 256) {    // stage A tile [16][256]
    const int row = i >> 8;
    const int c = i & 255;
    const int m = mtile * 16 + row;
    As[row * LDA + c] = (c < FEAT) ? selfF[(size_t)m * FEAT + c]
                                   : agg[(size_t)m * FEAT + (c - FEAT)];
  }
  __syncthreads();

  const int lane  = t & 31;
  const int ntile = t >> 5;                  // 0..7
  const int mloc  = lane & 15;               // A: M index per ISA layout
  const int koff  = (lane >> 4) * 2;         // A/B: K pair select per lane half
  const int n     = ntile * 16 + (lane & 15);

  v8f acc = {};
  for (int kb = 0; kb < K; kb += 4) {
    v2f a, bv;
    a.x  = As[mloc * LDA + kb + koff];
    a.y  = As[mloc * LDA + kb + koff + 1];
    bv.x = W[(size_t)(kb + koff) * EMBED + n];
    bv.y = W[(size_t)(kb + koff + 1) * EMBED + n];
    acc = __builtin_amdgcn_wmma_f32_16x16x4_f32(false, a, false, bv,
                                                (short)0, acc, false, false);
  }

  const int mrow = (lane >> 4) * 8;          // C/D: M = v + 8*(lane>=16)
#pragma unroll
  for (int v = 0; v < 8; ++v) {
    float x = acc[v];
    x = x > 0.0f ? x : 0.0f;
    const int m = mtile * 16 + mrow + v;
    E[(size_t)m * EMBED + n] = x;
  }
}

// ---------------------------------------------------------------------------
// Kernel 3: out = relu([self,E1,E2,E3] @ W_comb).T  -> [EMBED, B], K=512 WMMA,
// with an LDS transpose so stores of the transposed layout are contiguous.
// grid = B/16, block = 256.
// ---------------------------------------------------------------------------
__global__ void comb_gemm(const float* __restrict__ selfF,
                          const float* __restrict__ Ebase,
                          const float* __restrict__ Wc,
                          float* __restrict__ outT) {
  constexpr int K = FEAT + 3 * EMBED;        // 512
  constexpr int LDA = K + 4;
  __shared__ float As[16 * LDA];             // 33 KB
  __shared__ float Ct[8][16][17];            // per-wave 16x16 transpose tile

  const int mtile = blockIdx.x;
  const int t = threadIdx.x;
  for (int i = t; i < 16 * K; i += 256) {    // stage A tile [16][512]
    const int row = i >> 9;
    const int c = i & 511;
    const int m = mtile * 16 + row;
    const int seg = c >> 7;
    const int cc = c & 127;
    const float* __restrict__ src =
        (seg == 0) ? (selfF + (size_t)m * FEAT)
                   : (Ebase + ((size_t)(seg - 1) * B_NODES + m) * EMBED);
    As[row * LDA + c] = src[cc];
  }
  __syncthreads();

  const int lane  = t & 31;
  const int ntile = t >> 5;
  const int mloc  = lane & 15;
  const int koff  = (lane >> 4) * 2;
  const int n     = ntile * 16 + (lane & 15);

  v8f acc = {};
  for (int kb = 0; kb < K; kb += 4) {
    v2f a, bv;
    a.x  = As[mloc * LDA + kb + koff];
    a.y  = As[mloc * LDA + kb + koff + 1];
    bv.x = Wc[(size_t)(kb + koff) * EMBED + n];
    bv.y = Wc[(size_t)(kb + koff + 1) * EMBED + n];
    acc = __builtin_amdgcn_wmma_f32_16x16x4_f32(false, a, false, bv,
                                                (short)0, acc, false, false);
  }

  const int mrow = (lane >> 4) * 8;
#pragma unroll
  for (int v = 0; v < 8; ++v) {
    float x = acc[v];
    x = x > 0.0f ? x : 0.0f;
    Ct[ntile][mrow + v][lane & 15] = x;      // Ct[wave][m_local][n_local]
  }
  __syncthreads();

  // store transposed: out[n_g][m_g], contiguous over m (16 floats per half-wave)
  const int mcol = lane & 15;
  for (int rr = (lane >> 4); rr < 16; rr += 2) {
    const int n_g = ntile * 16 + rr;
    outT[(size_t)n_g * B_NODES + mtile * 16 + mcol] = Ct[ntile][mcol][rr];
  }
}

// ---------------------------------------------------------------------------
extern "C" void kernel_launch(void* const* d_in, const int* in_sizes, int n_in,
                              void* d_out, int out_size, void* d_ws, size_t ws_size,
                              hipStream_t stream) {
  (void)in_sizes; (void)n_in; (void)out_size; (void)ws_size;
  const float* feat  = (const float*)d_in[0];
  const float* Wclf  = (const float*)d_in[1];
  const float* bclf  = (const float*)d_in[2];
  const float* Wi1   = (const float*)d_in[3];
  const float* Wi2   = (const float*)d_in[4];
  const float* Wi3   = (const float*)d_in[5];
  const float* Wc    = (const float*)d_in[6];
  const int*   nodes = (const int*)d_in[7];
  const int*   n1    = (const int*)d_in[8];
  const int*   n2    = (const int*)d_in[9];
  const int*   n3    = (const int*)d_in[10];

  float* out    = (float*)d_out;
  float* outT   = out;                                   // [EMBED][B]
  float* scores = out + (size_t)EMBED * B_NODES;         // [B][2]

  float* ws    = (float*)d_ws;
  float* selfF = ws;                                     // B*FEAT
  float* cs0   = selfF + (size_t)B_NODES * FEAT;         // B
  float* aggB  = cs0 + B_NODES;                          // 3*B*FEAT
  float* Eb    = aggB + (size_t)3 * B_NODES * FEAT;      // 3*B*EMBED

  self_gather_scores<<<B_NODES, 128, 0, stream>>>(feat, nodes, Wclf, bclf,
                                                  selfF, cs0, scores);
  relation_agg<<<dim3(B_NODES, 3), 256, 0, stream>>>(feat, n1, n2, n3, Wclf,
                                                     bclf, cs0, aggB);
  intra_gemm<<<dim3(B_NODES / 16, 3), 256, 0, stream>>>(selfF, aggB, Wi1, Wi2,
                                                        Wi3, Eb);
  comb_gemm<<<B_NODES / 16, 256, 0, stream>>>(selfF, Eb, Wc, outT);
}